// GAT_22359599743214
// MI455X (gfx1250) — compile-verified
//
#include <hip/hip_runtime.h>
#include <math.h>

#define NN   50000
#define EE   800000
#define IND  256
#define H1_  4
#define HID1 64
#define OUTD 64
#define SLOPE 0.2f

typedef __attribute__((ext_vector_type(2))) float v2f;
typedef __attribute__((ext_vector_type(8))) float v8f;

// ---- monotonic ordered-uint encoding for float atomic max ----
__device__ __forceinline__ unsigned f2ord(float f) {
    unsigned u = __float_as_uint(f);
    return (u & 0x80000000u) ? ~u : (u | 0x80000000u);
}
__device__ __forceinline__ float ord2f(unsigned u) {
    return (u & 0x80000000u) ? __uint_as_float(u & 0x7fffffffu) : __uint_as_float(~u);
}

// =====================================================================
// WMMA f32 GEMM: C[M x NCOL] = A[M x K] * B[K x NCOL], row-major.
// Block = 256 threads = 8 waves. Each wave computes a 16x64 strip
// (four 16x16 accumulator tiles). BK = 32, K consumed 4 per WMMA.
// LDS tiles are stored k-pair interleaved: As[kpair][m][2] so each
// WMMA fragment (k, k+1) is one contiguous 8B ds_load_b64.
// (BM/16) * (BN/64) must equal 8.
// =====================================================================
template <int BM, int BN, int NCOL>
__global__ __launch_bounds__(256) void wmma_gemm_f32(
    const float* __restrict__ A, const float* __restrict__ B,
    float* __restrict__ C, int M, int K) {
    constexpr int BK  = 32;
    constexpr int BMp = BM + 2;   // pad: conflict-free interleaved writes
    constexpr int BNp = BN + 2;
    __shared__ float As[(BK / 2) * BMp * 2];   // As[kpair][m][sub]
    __shared__ float Bs[(BK / 2) * BNp * 2];   // Bs[kpair][n][sub]

    const int tid  = threadIdx.x;
    const int lane = tid & 31;
    const int wave = tid >> 5;
    constexpr int WCOLS = BN / 64;
    const int wrow = wave / WCOLS;          // row-wave index (16 rows each)
    const int wcol = wave % WCOLS;          // col-wave index (64 cols each)
    const int row0 = blockIdx.y * BM;
    const int col0 = blockIdx.x * BN;

    v8f acc[4];
#pragma unroll
    for (int t = 0; t < 4; t++)
#pragma unroll
        for (int r = 0; r < 8; r++) acc[t][r] = 0.0f;

    const int m16 = lane & 15;
    const int kb  = (lane >> 4) * 2;   // 0 for lanes 0-15, 2 for lanes 16-31

    const v2f* Asp = (const v2f*)As;
    const v2f* Bsp = (const v2f*)Bs;

    for (int k0 = 0; k0 < K; k0 += BK) {
        // cooperative tile loads (global reads coalesced on k / n)
        for (int idx = tid; idx < BM * BK; idx += 256) {
            int m = idx / BK, k = idx % BK;
            int r = row0 + m;
            float v = (r < M) ? A[(size_t)r * K + k0 + k] : 0.0f;
            As[((k >> 1) * BMp + m) * 2 + (k & 1)] = v;
        }
        for (int idx = tid; idx < BK * BN; idx += 256) {
            int k = idx / BN, n = idx % BN;
            Bs[((k >> 1) * BNp + n) * 2 + (k & 1)] =
                B[(size_t)(k0 + k) * NCOL + col0 + n];
        }
        __syncthreads();

        const int am = wrow * 16 + m16;
#pragma unroll
        for (int kk = 0; kk < BK; kk += 4) {
            const int kp = (kk + kb) >> 1;     // kpair index for this half-wave
            v2f a = Asp[kp * BMp + am];
#pragma unroll
            for (int t = 0; t < 4; t++) {
                v2f b = Bsp[kp * BNp + wcol * 64 + t * 16 + m16];
                acc[t] = __builtin_amdgcn_wmma_f32_16x16x4_f32(
                    false, a, false, b, (short)0, acc[t], false, false);
            }
        }
        __syncthreads();
    }

    // C/D layout: vgpr r -> row r (lanes 0-15) / row r+8 (lanes 16-31), col = lane&15
    const int rbase = row0 + wrow * 16 + ((lane >> 4) << 3);
#pragma unroll
    for (int t = 0; t < 4; t++) {
        int col = col0 + wcol * 64 + t * 16 + m16;
#pragma unroll
        for (int r = 0; r < 8; r++) {
            int row = rbase + r;
            if (row < M) C[(size_t)row * NCOL + col] = acc[t][r];
        }
    }
}

// ---- per-node attention scalars: el[n,h] = sum_d h[n,h,d]*al[h,d] ----
template <int H>
__global__ __launch_bounds__(256) void node_attn(
    const float* __restrict__ h, const float* __restrict__ al,
    const float* __restrict__ ar, float* __restrict__ el, float* __restrict__ er) {
    int lane = threadIdx.x & 31;
    int n = blockIdx.x * 8 + (threadIdx.x >> 5);
    if (n >= NN) return;
    const float* hp = h + (size_t)n * (H * 64);
#pragma unroll
    for (int hh = 0; hh < H; hh++) {
        float v0 = hp[hh * 64 + lane];
        float v1 = hp[hh * 64 + 32 + lane];
        float pl = v0 * al[hh * 64 + lane] + v1 * al[hh * 64 + 32 + lane];
        float pr = v0 * ar[hh * 64 + lane] + v1 * ar[hh * 64 + 32 + lane];
        for (int off = 16; off > 0; off >>= 1) {
            pl += __shfl_down(pl, off);
            pr += __shfl_down(pr, off);
        }
        if (lane == 0) {
            el[(size_t)n * H + hh] = pl;
            er[(size_t)n * H + hh] = pr;
        }
    }
}

// ---- edge pass 1: segment max of leaky_relu(el[src]+er[dst]) ----
template <int H>
__global__ void edge_max(const int* __restrict__ src, const int* __restrict__ dst,
                         const float* __restrict__ el, const float* __restrict__ er,
                         unsigned* __restrict__ emax) {
    int e = blockIdx.x * blockDim.x + threadIdx.x;
    if (e >= EE) return;
    int s = src[e], d = dst[e];
#pragma unroll
    for (int hh = 0; hh < H; hh++) {
        float sc = el[(size_t)s * H + hh] + er[(size_t)d * H + hh];
        sc = sc > 0.0f ? sc : SLOPE * sc;
        atomicMax(&emax[(size_t)d * H + hh], f2ord(sc));
    }
}

// ---- edge pass 2: ex = exp(e - emax[dst]); denom[dst] += ex ----
template <int H>
__global__ void edge_exp(const int* __restrict__ src, const int* __restrict__ dst,
                         const float* __restrict__ el, const float* __restrict__ er,
                         const unsigned* __restrict__ emax,
                         float* __restrict__ ex, float* __restrict__ denom) {
    int e = blockIdx.x * blockDim.x + threadIdx.x;
    if (e >= EE) return;
    int s = src[e], d = dst[e];
#pragma unroll
    for (int hh = 0; hh < H; hh++) {
        float sc = el[(size_t)s * H + hh] + er[(size_t)d * H + hh];
        sc = sc > 0.0f ? sc : SLOPE * sc;
        float m = ord2f(emax[(size_t)d * H + hh]);
        float v = expf(sc - m);
        ex[(size_t)e * H + hh] = v;
        atomicAdd(&denom[(size_t)d * H + hh], v);
    }
}

// ---- edge pass 3: out[dst] += (ex/denom[dst]) * h[src], one wave/edge ----
template <int H>
__global__ __launch_bounds__(256) void edge_aggregate(
    const int* __restrict__ src, const int* __restrict__ dst,
    const float* __restrict__ ex, const float* __restrict__ denom,
    const float* __restrict__ h, float* __restrict__ out) {
    int lane = threadIdx.x & 31;
    int e = blockIdx.x * 8 + (threadIdx.x >> 5);
    if (e >= EE) return;
    int s = src[e], d = dst[e];
    float a[H];
#pragma unroll
    for (int hh = 0; hh < H; hh++)
        a[hh] = ex[(size_t)e * H + hh] / denom[(size_t)d * H + hh];
    constexpr int T4 = (H * 64) / 4;
    const float4* hs = (const float4*)(h + (size_t)s * (H * 64));
    float* od = out + (size_t)d * (H * 64);
    for (int i = lane; i < T4; i += 32) {
        float4 v = hs[i];
        float c = a[(i * 4) >> 6];
        atomicAdd(&od[i * 4 + 0], c * v.x);
        atomicAdd(&od[i * 4 + 1], c * v.y);
        atomicAdd(&od[i * 4 + 2], c * v.z);
        atomicAdd(&od[i * 4 + 3], c * v.w);
    }
}

// ---- elementwise: x = elu(x + b)  (in place, cols = 256) ----
__global__ void elu_bias256(float* __restrict__ x, const float* __restrict__ b, int total) {
    int i = blockIdx.x * blockDim.x + threadIdx.x;
    if (i >= total) return;
    float v = x[i] + b[i & 255];
    x[i] = v > 0.0f ? v : expm1f(v);
}

// ---- init helpers ----
__global__ void fill_f32(float* p, float v, int n) {
    int i = blockIdx.x * blockDim.x + threadIdx.x;
    if (i < n) p[i] = v;
}
__global__ void fill_u32(unsigned* p, unsigned v, int n) {
    int i = blockIdx.x * blockDim.x + threadIdx.x;
    if (i < n) p[i] = v;
}
__global__ void init_out_bias(float* __restrict__ out, const float* __restrict__ b) {
    int i = blockIdx.x * blockDim.x + threadIdx.x;
    if (i < NN * OUTD) out[i] = b[i & 63];
}

extern "C" void kernel_launch(void* const* d_in, const int* in_sizes, int n_in,
                              void* d_out, int out_size, void* d_ws, size_t ws_size,
                              hipStream_t stream) {
    const float* feat = (const float*)d_in[0];
    const int*   src  = (const int*)d_in[1];
    const int*   dst  = (const int*)d_in[2];
    const float* W1   = (const float*)d_in[3];
    const float* al1  = (const float*)d_in[4];
    const float* ar1  = (const float*)d_in[5];
    const float* b1   = (const float*)d_in[6];
    const float* W2   = (const float*)d_in[7];
    const float* al2  = (const float*)d_in[8];
    const float* ar2  = (const float*)d_in[9];
    const float* b2   = (const float*)d_in[10];
    float* out = (float*)d_out;

    // ---- workspace carve-up ----
    float* h1    = (float*)d_ws;                 // N*256
    float* agg1  = h1    + (size_t)NN * 256;     // N*256 (becomes x2 after ELU)
    float* h2    = agg1  + (size_t)NN * 256;     // N*64
    float* el1   = h2    + (size_t)NN * 64;      // N*4
    float* er1   = el1   + (size_t)NN * H1_;
    float* den1  = er1   + (size_t)NN * H1_;
    unsigned* mx1 = (unsigned*)(den1 + (size_t)NN * H1_);  // N*4
    float* ex1   = (float*)(mx1 + (size_t)NN * H1_);       // E*4
    float* el2   = ex1   + (size_t)EE * H1_;     // N
    float* er2   = el2   + NN;
    float* den2  = er2   + NN;
    unsigned* mx2 = (unsigned*)(den2 + NN);      // N
    float* ex2   = (float*)(mx2 + NN);           // E

    const int TB = 256;
    // ---- init ----
    fill_f32<<<(NN * 256 + TB - 1) / TB, TB, 0, stream>>>(agg1, 0.0f, NN * 256);
    fill_f32<<<(NN * H1_ + TB - 1) / TB, TB, 0, stream>>>(den1, 0.0f, NN * H1_);
    fill_u32<<<(NN * H1_ + TB - 1) / TB, TB, 0, stream>>>(mx1, 0u, NN * H1_);
    fill_f32<<<(NN + TB - 1) / TB, TB, 0, stream>>>(den2, 0.0f, NN);
    fill_u32<<<(NN + TB - 1) / TB, TB, 0, stream>>>(mx2, 0u, NN);
    init_out_bias<<<(NN * OUTD + TB - 1) / TB, TB, 0, stream>>>(out, b2);

    // ================= Layer 1 =================
    // h1 = feat @ W1  (50000x256 @ 256x256)
    {
        dim3 grid(256 / 128, (NN + 63) / 64);
        wmma_gemm_f32<64, 128, 256><<<grid, 256, 0, stream>>>(feat, W1, h1, NN, IND);
    }
    node_attn<H1_><<<(NN + 7) / 8, 256, 0, stream>>>(h1, al1, ar1, el1, er1);
    edge_max<H1_><<<(EE + TB - 1) / TB, TB, 0, stream>>>(src, dst, el1, er1, mx1);
    edge_exp<H1_><<<(EE + TB - 1) / TB, TB, 0, stream>>>(src, dst, el1, er1, mx1, ex1, den1);
    edge_aggregate<H1_><<<(EE + 7) / 8, 256, 0, stream>>>(src, dst, ex1, den1, h1, agg1);
    // x2 = elu(agg1 + b1), in place
    elu_bias256<<<(NN * 256 + TB - 1) / TB, TB, 0, stream>>>(agg1, b1, NN * 256);

    // ================= Layer 2 =================
    // h2 = x2 @ W2  (50000x256 @ 256x64)
    {
        dim3 grid(64 / 64, (NN + 127) / 128);
        wmma_gemm_f32<128, 64, 64><<<grid, 256, 0, stream>>>(agg1, W2, h2, NN, 256);
    }
    node_attn<1><<<(NN + 7) / 8, 256, 0, stream>>>(h2, al2, ar2, el2, er2);
    edge_max<1><<<(EE + TB - 1) / TB, TB, 0, stream>>>(src, dst, el2, er2, mx2);
    edge_exp<1><<<(EE + TB - 1) / TB, TB, 0, stream>>>(src, dst, el2, er2, mx2, ex2, den2);
    // aggregate straight into d_out (pre-initialized with b2); H2=1 so head-mean is identity
    edge_aggregate<1><<<(EE + 7) / 8, 256, 0, stream>>>(src, dst, ex2, den2, h2, out);
}